// InverseCovMatrixFromPositions_19456201851267
// MI455X (gfx1250) — compile-verified
//
#include <hip/hip_runtime.h>

#define NN 20
#define WPB 8  // waves (batches) per block; 250000 = 8 * 31250 exactly

typedef float v2f __attribute__((ext_vector_type(2)));
typedef float v8f __attribute__((ext_vector_type(8)));

// Uniform broadcast from lane `srclane` (constant after unroll) via v_readlane -> SGPR.
__device__ __forceinline__ float bcastf(float v, int srclane) {
    return __int_as_float(__builtin_amdgcn_readlane(__float_as_int(v), srclane));
}

__global__ __launch_bounds__(WPB * 32)
void inv_cov20_kernel(const float* __restrict__ pos, float* __restrict__ out, int nb) {
    const int lane  = threadIdx.x & 31;
    const int wave  = threadIdx.x >> 5;
    const int batch = blockIdx.x * WPB + wave;
    if (batch >= nb) return;   // wave-uniform guard

    __shared__ __align__(16) float gmat[WPB][NN * NN];
    float* g = gmat[wave];

    // ---- load this batch's 20 points: lane l owns point l ----
    float x = 0.f, y = 0.f;
    if (lane < NN) {
        const float2 p = ((const float2*)pos)[(size_t)batch * NN + lane];
        x = p.x; y = p.y;
    }

    // ---- Gram matrix G = P * P^T via V_WMMA_F32_16X16X4_F32 (K padded 2->4) ----
    // 16x4 f32 A layout == 4x16 f32 B layout element-wise for this symmetric case:
    // lanes 0-15: {x_l, y_l}; lanes 16-31 (K=2,3 slots): zeros.
    v2f a00;                           // points 0..15
    a00.x = (lane < 16) ? x : 0.f;
    a00.y = (lane < 16) ? y : 0.f;

    const float xs = __shfl(x, lane + 16, 32);   // lane m -> x_{m+16}
    const float ys = __shfl(y, lane + 16, 32);
    v2f a10;                           // points 16..19 (rows 0..3), rest zero
    a10.x = (lane < 4) ? xs : 0.f;
    a10.y = (lane < 4) ? ys : 0.f;

    v8f c0 = {}; v8f c1 = {}; v8f c2 = {};
    // T00: rows 0-15 x cols 0-15
    c0 = __builtin_amdgcn_wmma_f32_16x16x4_f32(false, a00, false, a00, (short)0, c0, false, false);
    // T10: rows 16-19 x cols 0-15 (T01 = T10^T by symmetry)
    c1 = __builtin_amdgcn_wmma_f32_16x16x4_f32(false, a10, false, a00, (short)0, c1, false, false);
    // T11: rows 16-19 x cols 16-19
    c2 = __builtin_amdgcn_wmma_f32_16x16x4_f32(false, a10, false, a10, (short)0, c2, false, false);

    // ---- spill WMMA C layout (M split across lane halves) into LDS g[row][col] ----
    {
        const int col   = lane & 15;
        const int rbase = (lane >> 4) * 8;   // lanes 16-31 hold M = r+8
        #pragma unroll
        for (int r = 0; r < 8; ++r)
            g[(rbase + r) * NN + col] = c0[r];
        if (lane < 16) {
            #pragma unroll
            for (int r = 0; r < 4; ++r) {    // rows 16+r valid; M=4..7,8..15 are zero rows
                const float v = c1[r];
                g[(16 + r) * NN + lane] = v;         // G[16+r][n]
                g[lane * NN + (16 + r)] = v;         // G[n][16+r] (symmetry)
            }
        }
        if (lane < 4) {
            #pragma unroll
            for (int r = 0; r < 4; ++r)
                g[(16 + r) * NN + (16 + lane)] = c2[r];
        }
    }
    asm volatile("s_wait_dscnt 0" ::: "memory");  // LDS in-order per wave; fence compiler + hw

    // ---- lane i pulls row i of G (stride 20 floats -> 80B rows, float4-aligned) ----
    const int row = (lane < NN) ? lane : (NN - 1);
    float r[NN];
    {
        const float4* grow = (const float4*)(g + row * NN);
        #pragma unroll
        for (int c = 0; c < 5; ++c) {
            const float4 v = grow[c];
            r[4 * c + 0] = v.x; r[4 * c + 1] = v.y;
            r[4 * c + 2] = v.z; r[4 * c + 3] = v.w;
        }
    }

    // ---- cov row: sqd_ij = |p_i|^2 + |p_j|^2 - 2 G_ij ; cov = exp(-2*sqrt(sqd)), diag = 1.01 ----
    const float ni = x * x + y * y;
    #pragma unroll
    for (int j = 0; j < NN; ++j) {
        const float nj  = bcastf(ni, j);
        const float sqd = fmaxf(ni + nj - 2.f * r[j], 0.f);
        const float cv  = expf(-2.f * sqrtf(sqd));
        r[j] = (lane == j) ? 1.01f : cv;   // sigma^2*exp(0) + tau^2 exactly on diag
    }

    // ---- in-place Gauss-Jordan inverse (SPD: no pivoting), row-per-lane ----
    #pragma unroll
    for (int k = 0; k < NN; ++k) {
        float rk[NN];
        #pragma unroll
        for (int j = 0; j < NN; ++j) rk[j] = bcastf(r[j], k);  // SGPR pivot row
        const float ip = 1.0f / rk[k];
        if (lane == k) {
            #pragma unroll
            for (int j = 0; j < NN; ++j) r[j] = (j == k) ? ip : r[j] * ip;
        } else {
            const float f = r[k] * ip;
            #pragma unroll
            for (int j = 0; j < NN; ++j)
                r[j] = (j == k) ? -f : fmaf(-f, rk[j], r[j]);
        }
    }

    // ---- lane i stores row i of the inverse: 5 x b128, coalesced ----
    if (lane < NN) {
        float4* o = (float4*)(out + (size_t)batch * (NN * NN) + lane * NN);
        #pragma unroll
        for (int c = 0; c < 5; ++c) {
            const float4 v = { r[4 * c], r[4 * c + 1], r[4 * c + 2], r[4 * c + 3] };
            o[c] = v;
        }
    }
}

extern "C" void kernel_launch(void* const* d_in, const int* in_sizes, int n_in,
                              void* d_out, int out_size, void* d_ws, size_t ws_size,
                              hipStream_t stream) {
    (void)n_in; (void)out_size; (void)d_ws; (void)ws_size;
    const float* pos = (const float*)d_in[0];
    float* out       = (float*)d_out;
    const int nb     = in_sizes[0] / (NN * 2);        // 250000
    const int blocks = (nb + WPB - 1) / WPB;          // 31250, every wave full
    hipLaunchKernelGGL(inv_cov20_kernel, dim3(blocks), dim3(WPB * 32), 0, stream,
                       pos, out, nb);
}